// Net_55233279427136
// MI455X (gfx1250) — compile-verified
//
#include <hip/hip_runtime.h>

typedef __attribute__((ext_vector_type(2))) float v2f;
typedef __attribute__((ext_vector_type(8))) float v8f;

#define THRESH 1.0f
#define DECAY  0.01f

// -----------------------------------------------------------------------------
// 5x5 VALID conv, stride 1, implicit GEMM on V_WMMA_F32_16X16X4_F32.
// All dims are template constants -> every A/B access in the K loop is a
// global_load with a compile-time immediate offset (no address VALU, no LDS).
//
// K-space is split between the two half-waves BY CHANNEL (reduction order is
// free): lanes 0-15 reduce channels [0, CIN/2), lanes 16-31 reduce
// [CIN/2, CIN). The im2col offset is linear in channel, so the half-wave shift
// is one constant folded into each lane's base pointer.
//   Slot (half h, reg j) at step kh consumes k = h*KH + kh + j (KH = K/2).
//   A (16x4 f32): lane (l16,h) reg j = wgt[l16-th row of m-tile][h*KH + kh + j]
//   B (4x16 f32): lane (l16,h) reg j = im2col[h*KH + kh + j][ncol]
//   C/D (16x16 f32): vgpr r, lane L -> M = r + (L>=16?8:0), N = L&15.
//
// MT m-tiles x NT n-tiles per wave: each A fragment feeds NT WMMAs and each B
// fragment feeds MT WMMAs (MT*NT independent accumulator chains).
// Out-of-range N lanes read a zero-filled dummy buffer (true zeros, no masks);
// EXEC stays all-ones around every WMMA (ISA requirement).
// Epilogue fuses the LIF membrane accumulate: mem += conv + bias - DECAY.
// -----------------------------------------------------------------------------
template<int MT, int NT, int CIN, int HIN, int WIN, int HOUT, int WOUT>
__global__ __launch_bounds__(32) void conv5x5_wmma_lif(
    const float* __restrict__ in,    // [B, CIN, HIN, WIN] (base already offset for t)
    const float* __restrict__ wgt,   // [COUT, CIN, 5, 5] == [COUT, K]
    const float* __restrict__ bias,  // [COUT]
    float*       __restrict__ mem,   // [B, COUT, HOUT, WOUT]  (accumulated in place)
    const float* __restrict__ zbuf,  // zero-filled dummy (>= (CIN/2)*HIN*WIN floats)
    long inBatchStride)
{
    constexpr int K      = CIN * 25;
    constexpr int KH     = K / 2;            // K handled per half-wave
    constexpr int COUT   = MT * 16;
    constexpr int N      = HOUT * WOUT;
    constexpr int HW     = HIN * WIN;
    constexpr int CHUNKS = KH / 50;          // 2 channels (50 k) per chunk per half
    constexpr int REMK   = KH - CHUNKS * 50; // conv1: 25, conv2: 0

    const int lane  = threadIdx.x;           // 0..31
    const int half  = lane >> 4;
    const int l16   = lane & 15;
    const int b     = blockIdx.z;
    const int nbase = blockIdx.x * (16 * NT);

    v8f acc[MT * NT];
#pragma unroll
    for (int mt = 0; mt < MT; ++mt)
#pragma unroll
        for (int nt = 0; nt < NT; ++nt)
#pragma unroll
            for (int r = 0; r < 8; ++r)
                acc[mt * NT + nt][r] = bias[mt * 16 + r + half * 8];

    // Per-n-tile lane columns; invalid columns read the zero dummy buffer.
    int         ncol[NT];
    bool        nval[NT];
    const float* pB[NT];
#pragma unroll
    for (int nt = 0; nt < NT; ++nt) {
        const int n = nbase + nt * 16 + l16;
        ncol[nt] = n;
        nval[nt] = (n < N);
        const int oy = n / WOUT;             // harmless garbage if invalid
        const int ox = n - oy * WOUT;
        pB[nt] = nval[nt]
            ? (in + (size_t)b * inBatchStride + (size_t)oy * WIN + ox + half * (CIN / 2) * HW)
            : zbuf;
    }
    const float* pA = wgt + (size_t)l16 * K + half * KH;

    for (int ch = 0; ch < CHUNKS; ++ch) {
        const float* pAc = pA + ch * 50;
        const float* pBc[NT];
#pragma unroll
        for (int nt = 0; nt < NT; ++nt) pBc[nt] = pB[nt] + ch * 2 * HW;

#pragma unroll
        for (int s = 0; s < 25; ++s) {
            const int kh0 = 2 * s, kh1 = 2 * s + 1;
            const int c0 = kh0 / 25, r0 = kh0 - c0 * 25;
            const int c1 = kh1 / 25, r1 = kh1 - c1 * 25;
            v2f bb[NT];
#pragma unroll
            for (int nt = 0; nt < NT; ++nt) {
                bb[nt].x = pBc[nt][c0 * HW + (r0 / 5) * WIN + (r0 % 5)];
                bb[nt].y = pBc[nt][c1 * HW + (r1 / 5) * WIN + (r1 % 5)];
            }
#pragma unroll
            for (int mt = 0; mt < MT; ++mt) {
                v2f a;
                if constexpr ((KH & 1) == 0) {      // b64, 8B-aligned
                    a = *(const v2f*)(pAc + mt * 16 * K + kh0);
                } else {
                    a.x = pAc[mt * 16 * K + kh0];
                    a.y = pAc[mt * 16 * K + kh1];
                }
#pragma unroll
                for (int nt = 0; nt < NT; ++nt)
                    acc[mt * NT + nt] = __builtin_amdgcn_wmma_f32_16x16x4_f32(
                        false, a, false, bb[nt], (short)0, acc[mt * NT + nt], false, false);
            }
        }
    }

    if constexpr (REMK > 0) {                // conv1: 25 k per half, last slot padded
        const float* pAt = pA + CHUNKS * 50;
        const float* pBt[NT];
#pragma unroll
        for (int nt = 0; nt < NT; ++nt) pBt[nt] = pB[nt] + CHUNKS * 2 * HW;

#pragma unroll
        for (int s = 0; s < (REMK + 1) / 2; ++s) {
            const int kh0 = 2 * s, kh1 = 2 * s + 1;
            const int c0 = kh0 / 25, r0 = kh0 - c0 * 25;
            const int c1 = kh1 / 25, r1 = kh1 - c1 * 25;
            v2f bb[NT];
#pragma unroll
            for (int nt = 0; nt < NT; ++nt) {
                bb[nt].x = pBt[nt][c0 * HW + (r0 / 5) * WIN + (r0 % 5)];
                bb[nt].y = (kh1 < REMK) ? pBt[nt][c1 * HW + (r1 / 5) * WIN + (r1 % 5)] : 0.0f;
            }
#pragma unroll
            for (int mt = 0; mt < MT; ++mt) {
                v2f a;
                a.x = pAt[mt * 16 * K + kh0];
                a.y = (kh1 < REMK) ? pAt[mt * 16 * K + kh1] : 0.0f;
#pragma unroll
                for (int nt = 0; nt < NT; ++nt)
                    acc[mt * NT + nt] = __builtin_amdgcn_wmma_f32_16x16x4_f32(
                        false, a, false, bb[nt], (short)0, acc[mt * NT + nt], false, false);
            }
        }
    }

    // Epilogue: mem[b][o][n] = mem_old - DECAY + conv  (each element uniquely owned).
#pragma unroll
    for (int nt = 0; nt < NT; ++nt) {
        if (nval[nt]) {
            float* memB = mem + ((size_t)b * COUT) * N + ncol[nt];
#pragma unroll
            for (int mt = 0; mt < MT; ++mt)
#pragma unroll
                for (int r = 0; r < 8; ++r) {
                    const size_t off = (size_t)(mt * 16 + r + half * 8) * N;
                    memB[off] = memB[off] - DECAY + acc[mt * NT + nt][r];
                }
        }
    }
}

// -----------------------------------------------------------------------------
// LIF stage 1: p1 = sumpool2x2(mem1_new); mem1 = (mem1_new >= THRESH) ? 0 : mem1_new
// -----------------------------------------------------------------------------
__global__ void lif_pool_kernel(float* __restrict__ mem, float* __restrict__ pooled,
                                int npooled, int Hc, int Wc, int Wp, int HpWp)
{
    const int idx = blockIdx.x * blockDim.x + threadIdx.x;
    if (idx >= npooled) return;
    const int bc = idx / HpWp;
    const int r  = idx - bc * HpWp;
    const int py = r / Wp;
    const int px = r - py * Wp;
    const size_t cbase = (size_t)bc * Hc * Wc;
    float s = 0.0f;
#pragma unroll
    for (int dy = 0; dy < 2; ++dy)
#pragma unroll
        for (int dx = 0; dx < 2; ++dx) {
            const size_t off = cbase + (size_t)(2 * py + dy) * Wc + (2 * px + dx);
            const float m = mem[off];
            s += m;
            mem[off] = (m >= THRESH) ? 0.0f : m;
        }
    pooled[idx] = s;
}

// -----------------------------------------------------------------------------
// LIF stage 2: spk[b][o] += sum_{H,W} mem2_new ; reset mem2.
// (pool2x2 + spatial sum == full spatial sum since 26 = 13*2)
// -----------------------------------------------------------------------------
__global__ __launch_bounds__(256) void lif_spk_kernel(float* __restrict__ mem,
                                                      float* __restrict__ spk,
                                                      int spatial)
{
    __shared__ float red[256];
    const int bo = blockIdx.x;
    const size_t base = (size_t)bo * spatial;
    float s = 0.0f;
    for (int i = threadIdx.x; i < spatial; i += 256) {
        const float m = mem[base + i];
        s += m;
        mem[base + i] = (m >= THRESH) ? 0.0f : m;
    }
    red[threadIdx.x] = s;
    __syncthreads();
    for (int st = 128; st > 0; st >>= 1) {
        if (threadIdx.x < st) red[threadIdx.x] += red[threadIdx.x + st];
        __syncthreads();
    }
    if (threadIdx.x == 0) spk[bo] += red[0];
}

// -----------------------------------------------------------------------------
// Final FC: out[b][o] = spk[b] . fc_w[o] + fc_b[o]   (64x64 @ 64x10 -> tiny)
// -----------------------------------------------------------------------------
__global__ void fc_kernel(const float* __restrict__ spk, const float* __restrict__ w,
                          const float* __restrict__ bias, float* __restrict__ out)
{
    const int idx = blockIdx.x * blockDim.x + threadIdx.x;
    if (idx >= 64 * 10) return;
    const int b = idx / 10;
    const int o = idx - b * 10;
    float s = bias[o];
#pragma unroll 8
    for (int cc = 0; cc < 64; ++cc) s += spk[b * 64 + cc] * w[o * 64 + cc];
    out[idx] = s;
}

__global__ void zero_kernel(float* __restrict__ p, size_t n)
{
    const size_t i = (size_t)blockIdx.x * blockDim.x + threadIdx.x;
    if (i < n) p[i] = 0.0f;
}

// -----------------------------------------------------------------------------
extern "C" void kernel_launch(void* const* d_in, const int* in_sizes, int n_in,
                              void* d_out, int out_size, void* d_ws, size_t ws_size,
                              hipStream_t stream)
{
    const float* x  = (const float*)d_in[0]; // [64,25,2,64,64]
    const float* w1 = (const float*)d_in[1]; // [32,2,5,5]
    const float* b1 = (const float*)d_in[2]; // [32]
    const float* w2 = (const float*)d_in[3]; // [64,32,5,5]
    const float* b2 = (const float*)d_in[4]; // [64]
    const float* fw = (const float*)d_in[5]; // [10,64]
    const float* fb = (const float*)d_in[6]; // [10]
    float* out = (float*)d_out;              // [64,10]

    // Workspace layout (floats). Zeroed region first (states + spk + zero dummy).
    float* ws = (float*)d_ws;
    size_t o = 0;
    float* mem1 = ws + o; o += (size_t)64 * 32 * 60 * 60;  // 7,372,800
    float* mem2 = ws + o; o += (size_t)64 * 64 * 26 * 26;  // 2,768,896
    float* spk  = ws + o; o += 64 * 64;                    //     4,096
    float* zbuf = ws + o; o += 16 * 30 * 30;               //    14,400 (stays zero)
    const size_t nzero = o;
    float* p1   = ws + o; o += (size_t)64 * 32 * 30 * 30;  // 1,843,200
    // total ~ 48 MB

    zero_kernel<<<(unsigned)((nzero + 255) / 256), 256, 0, stream>>>(ws, nzero);

    const int np1 = 64 * 32 * 30 * 30;

    for (int t = 0; t < 25; ++t) {
        const float* xt = x + (size_t)t * 2 * 64 * 64;

        // conv1 + mem1 accumulate: M=32 (MT=2), N=3600 (113 tiles of 32), K=50
        conv5x5_wmma_lif<2, 2, 2, 64, 64, 60, 60><<<dim3(113, 1, 64), 32, 0, stream>>>(
            xt, w1, b1, mem1, zbuf, /*inBatchStride*/(long)25 * 2 * 64 * 64);

        // pool 2x2 of mem1 -> p1; threshold-reset mem1
        lif_pool_kernel<<<(np1 + 255) / 256, 256, 0, stream>>>(
            mem1, p1, np1, /*Hc*/60, /*Wc*/60, /*Wp*/30, /*HpWp*/900);

        // conv2 + mem2 accumulate: M=64 (MT=4), N=676 (22 tiles of 32), K=800
        conv5x5_wmma_lif<4, 2, 32, 30, 30, 26, 26><<<dim3(22, 1, 64), 32, 0, stream>>>(
            p1, w2, b2, mem2, zbuf, /*inBatchStride*/(long)32 * 30 * 30);

        // spatial sum of mem2 into spk accumulator; threshold-reset mem2
        lif_spk_kernel<<<64 * 64, 256, 0, stream>>>(mem2, spk, 26 * 26);
    }

    fc_kernel<<<3, 256, 0, stream>>>(spk, fw, fb, out);
}